// UniformNeighborSampler_83021717832676
// MI455X (gfx1250) — compile-verified
//
#include <hip/hip_runtime.h>
#include <hip/hip_bf16.h>
#include <stdint.h>

// UniformNeighborSampler: out[i][j] = adj_info[ ids[i] ][ randpicks[i][j] ]
// MI455X (gfx1250) strategy (hot path, sampler_fixed<10>):
//   - Each wave32 owns IDS_PER_WAVE = 16 adjacency rows (512B each).
//   - One coalesced load fetches the wave's 16 node ids; v_readlane scalarizes
//     each node id so async-copy addresses are scalar-base + lane*16.
//   - 16x GLOBAL_LOAD_ASYNC_TO_LDS_B128: lane l moves bytes [16l,16l+16) of a
//     row -> one instruction stages one full 512B row into the wave's LDS slice.
//   - s_wait_asynccnt 0, then 5 fully-unrolled iterations resolve the random
//     picks from LDS with linear (coalesced) picks loads and out stores.
// Roofline: ~13 MB total HBM traffic @ 23.3 TB/s -> latency-bound; 1600 waves
// each holding 16 async copies in flight hides it. No matrix math exists in
// this op -> wmma correctly stays 0; the CDNA5 feature that pays here is the
// async global->LDS DMA path (512B sequential bursts instead of ~4 scattered
// 128B lines per row, same bytes, cleaner HBM access pattern).

#define MAX_DEGREE      128   // row length of adj_info (fixed by reference setup)
#define IDS_PER_WAVE    16
#define WAVES_PER_BLOCK 8
#define IDS_PER_BLOCK   (IDS_PER_WAVE * WAVES_PER_BLOCK)   // 128
#define BLOCK_THREADS   (WAVES_PER_BLOCK * 32)             // 256

typedef int v4i __attribute__((vector_size(4 * sizeof(int))));
typedef __attribute__((address_space(1))) v4i* global_v4i_ptr;
typedef __attribute__((address_space(3))) v4i* lds_v4i_ptr;

__device__ __forceinline__ void async_copy_b128_to_lds(const int* gsrc, int* ldst) {
#if defined(__gfx1250__) && __has_builtin(__builtin_amdgcn_global_load_async_to_lds_b128)
    __builtin_amdgcn_global_load_async_to_lds_b128(
        (global_v4i_ptr)(unsigned long long)(size_t)gsrc,
        (lds_v4i_ptr)(unsigned int)(size_t)ldst,     // low 32 bits of generic ptr == LDS offset
        /*offset=*/0, /*cpol=*/0);
#else
    unsigned int lds_off = (unsigned int)(size_t)ldst;
    asm volatile("global_load_async_to_lds_b128 %0, %1, off"
                 :: "v"(lds_off), "v"(gsrc)
                 : "memory");
#endif
}

__device__ __forceinline__ void wait_asynccnt0() {
#if defined(__gfx1250__) && __has_builtin(__builtin_amdgcn_s_wait_asynccnt)
    __builtin_amdgcn_s_wait_asynccnt(0);
#else
    asm volatile("s_wait_asynccnt 0" ::: "memory");
#endif
    asm volatile("" ::: "memory");   // keep LDS reads after the wait
}

// ---------------------------------------------------------------------------
// Hot path: num_samples == pick_stride == NS known at compile time.
// picks/out indices are then linear in s -> fully coalesced, fully unrolled.
// ---------------------------------------------------------------------------
template <int NS>
__global__ void __launch_bounds__(BLOCK_THREADS)
sampler_fixed(const int* __restrict__ adj,
              const int* __restrict__ ids,
              const int* __restrict__ picks,
              int* __restrict__ out,
              int n_ids) {
    __shared__ int rows[IDS_PER_BLOCK * MAX_DEGREE];   // 64 KB of 320 KB WGP LDS

    const int wave    = (int)threadIdx.x / 32;
    const int lane    = (int)threadIdx.x % 32;
    const int id_base = (int)blockIdx.x * IDS_PER_BLOCK + wave * IDS_PER_WAVE;

    int* wave_lds = &rows[(wave * IDS_PER_WAVE) * MAX_DEGREE];

    // One coalesced load of this wave's 16 node ids (lanes 16..31 replicate 0..15).
    int q = id_base + (lane & (IDS_PER_WAVE - 1));
    q = q < n_ids ? q : n_ids - 1;
    const int node_lane = ids[q];

    // ---- Phase 1: stage 16 full rows into LDS via the async DMA path ----
#pragma unroll
    for (int r = 0; r < IDS_PER_WAVE; ++r) {
        int node = __builtin_amdgcn_readlane(node_lane, r);   // wave-uniform scalar
        const int* src = adj + (long long)node * MAX_DEGREE + lane * 4;
        int*       dst = wave_lds + r * MAX_DEGREE + lane * 4;
        async_copy_b128_to_lds(src, dst);
    }
    wait_asynccnt0();
    // Each wave reads only its own LDS slice -> no workgroup barrier needed.

    // ---- Phase 2: resolve picks from LDS; picks/out are linear in s ----
    const long long base = (long long)id_base * NS;
    constexpr int TOTAL = IDS_PER_WAVE * NS;              // 160 for NS=10
#pragma unroll
    for (int k = 0; k < (TOTAL + 31) / 32; ++k) {
        int s = lane + 32 * k;
        if (s < TOTAL && id_base + s / NS < n_ids) {
            int p = picks[base + s] & (MAX_DEGREE - 1);
            out[base + s] = wave_lds[(s / NS) * MAX_DEGREE + p];
        }
    }
}

// ---------------------------------------------------------------------------
// Tiny generic fallback: direct per-sample gather (only used if the shapes
// ever deviate from num_samples == pick_stride == 10; never hot).
// ---------------------------------------------------------------------------
__global__ void __launch_bounds__(256)
sampler_direct(const int* __restrict__ adj,
               const int* __restrict__ ids,
               const int* __restrict__ picks,
               int* __restrict__ out,
               int num_samples, int pick_stride, int total) {
    int t = (int)blockIdx.x * 256 + (int)threadIdx.x;
    if (t >= total) return;
    int i = t / num_samples;
    int j = t - i * num_samples;
    int node = ids[i];
    int p    = picks[(long long)i * pick_stride + j];
    out[t] = adj[(long long)node * MAX_DEGREE + p];
}

extern "C" void kernel_launch(void* const* d_in, const int* in_sizes, int n_in,
                              void* d_out, int out_size, void* d_ws, size_t ws_size,
                              hipStream_t stream) {
    const int* adj   = (const int*)d_in[0];   // [N_NODES, MAX_DEGREE] int32
    const int* ids   = (const int*)d_in[1];   // [N_IDS] int32
    const int* picks = (const int*)d_in[2];   // [N_IDS, NUM_SAMPLES] int32
    // d_in[3] is the device-resident num_samples scalar; derive on host instead:
    const int n_ids       = in_sizes[1];
    const int num_samples = out_size / n_ids;        // 10
    const int pick_stride = in_sizes[2] / n_ids;     // 10

    int* out = (int*)d_out;

    if (num_samples == 10 && pick_stride == 10) {
        const int blocks = (n_ids + IDS_PER_BLOCK - 1) / IDS_PER_BLOCK;  // 200
        sampler_fixed<10><<<dim3(blocks), dim3(BLOCK_THREADS), 0, stream>>>(
            adj, ids, picks, out, n_ids);
    } else {
        const int total  = n_ids * num_samples;
        const int blocks = (total + 255) / 256;
        sampler_direct<<<dim3(blocks), dim3(256), 0, stream>>>(
            adj, ids, picks, out, num_samples, pick_stride, total);
    }
}